// LatentSpaceNSVQ_10771777978777
// MI455X (gfx1250) — compile-verified
//
#include <hip/hip_runtime.h>
#include <math.h>

typedef __attribute__((ext_vector_type(2))) float v2f;
typedef __attribute__((ext_vector_type(8))) float v8f;

#define BB   32
#define HH   32
#define WW   32
#define DIN  1024
#define DE   64
#define KCB  8192
#define SS   16
#define NROWS 512          // B*S
#define EPSF 1e-12f

// ---- workspace layout (float offsets) ----
#define WS_POOLED   0                        // 512*1024
#define WS_FLAT     (512*1024)               // 512*64
#define WS_QUANT    (WS_FLAT + 512*64)       // 512*64
#define WS_Y        (WS_QUANT + 512*64)      // 512*1024
#define WS_CNORM    (WS_Y + 512*1024)        // 8192
#define WS_COUNTS   (WS_CNORM + 8192)        // 8192 (uint)
#define WS_MINIDX   (WS_COUNTS + 8192)       // 512 (int)

// ---- output layout (float elements) ----
#define OUT_DECODED 0
#define OUT_PERP    (BB*HH*WW*DIN)           // 33554432
#define OUT_USED    (OUT_PERP + 1)
#define OUT_IDX     (OUT_USED + KCB)

// ============================================================
// K1: pooled[b,g,c] = mean over 8x8 block of (t1 - t0). 268MB stream.
// ============================================================
__global__ __launch_bounds__(256) void k_pool(const float* __restrict__ t0,
                                              const float* __restrict__ t1,
                                              float* __restrict__ pooled) {
  const int cell = blockIdx.x;          // 0..15 (gh*4+gw)
  const int b    = blockIdx.y;          // 0..31
  const int gh = cell >> 2, gw = cell & 3;
  const int c4 = threadIdx.x;           // channel group: 4 floats
  const float4* __restrict__ a4 = (const float4*)t0;
  const float4* __restrict__ b4 = (const float4*)t1;
  float sx = 0.f, sy = 0.f, sz = 0.f, sw = 0.f;
  for (int p = 0; p < 64; ++p) {
    int py = p >> 3, px = p & 7;
    size_t pix = (size_t)((b * HH + gh * 8 + py) * WW + gw * 8 + px);
    size_t idx = pix * (DIN / 4) + c4;
    float4 x0 = a4[idx];
    float4 x1 = b4[idx];
    sx += x1.x - x0.x; sy += x1.y - x0.y;
    sz += x1.z - x0.z; sw += x1.w - x0.w;
  }
  float4 o; const float inv = 1.0f / 64.0f;
  o.x = sx * inv; o.y = sy * inv; o.z = sz * inv; o.w = sw * inv;
  ((float4*)pooled)[(size_t)(b * 16 + cell) * (DIN / 4) + c4] = o;
}

// ============================================================
// K2: flat = pooled @ W_in^T + b_in   (M=512,N=64,K=1024), WMMA f32 16x16x4
// 128 tiles, one wave each.
// ============================================================
__global__ __launch_bounds__(32) void k_encode(const float* __restrict__ pooled,
                                               const float* __restrict__ Win,
                                               const float* __restrict__ bin,
                                               float* __restrict__ flat) {
  const int mtile = blockIdx.x >> 2;    // 0..31
  const int ntile = blockIdx.x & 3;     // 0..3
  const int lane  = threadIdx.x;
  const int loc   = lane & 15;
  const int koff  = (lane >> 4) << 1;   // 0 or 2
  const float* __restrict__ arow = pooled + (size_t)((mtile << 4) + loc) * DIN;
  const int ncol = (ntile << 4) + loc;
  const float* __restrict__ brow = Win + (size_t)ncol * DIN;   // B[k][n]=W_in[n][k]
  v8f acc = {0.f,0.f,0.f,0.f,0.f,0.f,0.f,0.f};
  #pragma unroll 8
  for (int kb = 0; kb < DIN; kb += 4) {
    v2f a = *(const v2f*)(arow + kb + koff);
    v2f b = *(const v2f*)(brow + kb + koff);
    acc = __builtin_amdgcn_wmma_f32_16x16x4_f32(false, a, false, b,
                                                (short)0, acc, false, false);
  }
  const float bias = bin[ncol];
  const int mbase = (mtile << 4) + ((lane >> 4) << 3);
  #pragma unroll
  for (int i = 0; i < 8; ++i)
    flat[(size_t)(mbase + i) * DE + ncol] = acc[i] + bias;
}

// ============================================================
// K2.5: cnorm[k] = ||codebook_k||^2 ; zero counts
// ============================================================
__global__ __launch_bounds__(256) void k_cnorm(const float* __restrict__ cb,
                                               float* __restrict__ cnorm,
                                               unsigned* __restrict__ counts) {
  int n = blockIdx.x * 256 + threadIdx.x;   // 8192
  const float4* __restrict__ r = (const float4*)(cb + (size_t)n * DE);
  float s = 0.f;
  #pragma unroll
  for (int j = 0; j < DE / 4; ++j) {
    float4 v = r[j];
    s += v.x * v.x + v.y * v.y + v.z * v.z + v.w * v.w;
  }
  cnorm[n] = s;
  counts[n] = 0u;
}

// ============================================================
// K3: nearest codebook: argmin_k ( ||c_k||^2 - 2 f.c_k ). WMMA GEMM
// M=512, N=8192, K=64. One workgroup per 16-row M-tile; 8 waves sweep
// the 512 N-tiles; A tile held in registers for the whole sweep.
// ============================================================
__global__ __launch_bounds__(256) void k_argmin(const float* __restrict__ flat,
                                                const float* __restrict__ cb,
                                                const float* __restrict__ cnorm,
                                                unsigned* __restrict__ counts,
                                                int* __restrict__ minidx,
                                                int* __restrict__ idx_out) {
  __shared__ float lmin[8][16];
  __shared__ int   lidx[8][16];
  const int mtile = blockIdx.x;         // 0..31
  const int wid   = threadIdx.x >> 5;   // 0..7
  const int lane  = threadIdx.x & 31;
  const int loc   = lane & 15;
  const int koff  = (lane >> 4) << 1;

  const float* __restrict__ arow = flat + (size_t)((mtile << 4) + loc) * DE;
  v2f a[16];
  #pragma unroll
  for (int j = 0; j < 16; ++j) a[j] = *(const v2f*)(arow + j * 4 + koff);

  float minv[8]; int mini[8];
  #pragma unroll
  for (int i = 0; i < 8; ++i) { minv[i] = 3.4e38f; mini[i] = 0; }

  for (int t = wid; t < KCB / 16; t += 8) {
    const int nbase = t << 4;
    const float* __restrict__ brow = cb + (size_t)(nbase + loc) * DE; // B[k][n]=cb[n][k]
    v8f acc = {0.f,0.f,0.f,0.f,0.f,0.f,0.f,0.f};
    #pragma unroll
    for (int j = 0; j < 16; ++j) {
      v2f b = *(const v2f*)(brow + j * 4 + koff);
      acc = __builtin_amdgcn_wmma_f32_16x16x4_f32(false, a[j], false, b,
                                                  (short)0, acc, false, false);
    }
    const int n = nbase + loc;
    const float cn = cnorm[n];
    #pragma unroll
    for (int i = 0; i < 8; ++i) {
      float sc = cn - 2.0f * acc[i];
      if (sc < minv[i] || (sc == minv[i] && n < mini[i])) { minv[i] = sc; mini[i] = n; }
    }
  }
  // reduce across the 16 lanes sharing each m (xor stays within the half)
  #pragma unroll
  for (int i = 0; i < 8; ++i) {
    #pragma unroll
    for (int off = 1; off < 16; off <<= 1) {
      float ov = __shfl_xor(minv[i], off, 32);
      int   oi = __shfl_xor(mini[i], off, 32);
      if (ov < minv[i] || (ov == minv[i] && oi < mini[i])) { minv[i] = ov; mini[i] = oi; }
    }
    if (loc == 0) {
      int m = ((lane >> 4) << 3) + i;
      lmin[wid][m] = minv[i];
      lidx[wid][m] = mini[i];
    }
  }
  __syncthreads();
  if (threadIdx.x < 16) {
    int m = threadIdx.x;
    float bv = lmin[0][m]; int bi = lidx[0][m];
    #pragma unroll
    for (int w = 1; w < 8; ++w) {
      float ov = lmin[w][m]; int oi = lidx[w][m];
      if (ov < bv || (ov == bv && oi < bi)) { bv = ov; bi = oi; }
    }
    int row = (mtile << 4) + m;
    minidx[row] = bi;
    idx_out[row] = bi;
    atomicAdd(&counts[bi], 1u);
  }
}

// ============================================================
// K4: NSVQ: quantized = flat + (||flat-hard|| / (||noise||+eps)) * noise
// One wave per row (DE=64 -> 2 elements/lane).
// ============================================================
__global__ __launch_bounds__(32) void k_quant(const float* __restrict__ flat,
                                              const float* __restrict__ cb,
                                              const float* __restrict__ noise,
                                              const int* __restrict__ minidx,
                                              float* __restrict__ quant) {
  const int n = blockIdx.x;            // 0..511
  const int lane = threadIdx.x;        // 0..31
  const int idx = minidx[n];
  const float* __restrict__ f = flat  + (size_t)n * DE;
  const float* __restrict__ c = cb    + (size_t)idx * DE;
  const float* __restrict__ z = noise + (size_t)n * DE;
  float f0 = f[lane], f1 = f[lane + 32];
  float d0 = f0 - c[lane], d1 = f1 - c[lane + 32];
  float z0 = z[lane], z1 = z[lane + 32];
  float rs = d0 * d0 + d1 * d1;
  float ns = z0 * z0 + z1 * z1;
  #pragma unroll
  for (int off = 1; off < 32; off <<= 1) {
    rs += __shfl_xor(rs, off, 32);
    ns += __shfl_xor(ns, off, 32);
  }
  float scale = sqrtf(rs) / (sqrtf(ns) + EPSF);
  quant[(size_t)n * DE + lane]      = f0 + scale * z0;
  quant[(size_t)n * DE + lane + 32] = f1 + scale * z1;
}

// ============================================================
// K5: perplexity + codebooks_used
// ============================================================
__global__ __launch_bounds__(256) void k_perp(const unsigned* __restrict__ counts,
                                              float* __restrict__ out_perp,
                                              int* __restrict__ out_used) {
  __shared__ float sh[256];
  float acc = 0.f;
  for (int b = threadIdx.x; b < KCB; b += 256) {
    unsigned c = counts[b];
    out_used[b] = (int)c;
    float p = (float)c * (1.0f / (float)NROWS);
    acc += p * logf(p + EPSF);
  }
  sh[threadIdx.x] = acc;
  __syncthreads();
  for (int s = 128; s > 0; s >>= 1) {
    if (threadIdx.x < s) sh[threadIdx.x] += sh[threadIdx.x + s];
    __syncthreads();
  }
  if (threadIdx.x == 0) *out_perp = expf(-sh[0]);
}

// ============================================================
// K6: Y = quantized @ W_out^T + b_out  (M=512,N=1024,K=64), WMMA f32.
// b_out folded here is exact: bilinear weights sum to 1.
// ============================================================
__global__ __launch_bounds__(32) void k_decode_gemm(const float* __restrict__ quant,
                                                    const float* __restrict__ Wout,
                                                    const float* __restrict__ bout,
                                                    float* __restrict__ Y) {
  const int mtile = blockIdx.x >> 6;    // 0..31
  const int ntile = blockIdx.x & 63;    // 0..63
  const int lane  = threadIdx.x;
  const int loc   = lane & 15;
  const int koff  = (lane >> 4) << 1;
  const float* __restrict__ arow = quant + (size_t)((mtile << 4) + loc) * DE;
  const int ncol = (ntile << 4) + loc;
  const float* __restrict__ brow = Wout + (size_t)ncol * DE;  // B[d][c]=W_out[c][d]
  v8f acc = {0.f,0.f,0.f,0.f,0.f,0.f,0.f,0.f};
  #pragma unroll
  for (int j = 0; j < 16; ++j) {
    v2f a = *(const v2f*)(arow + j * 4 + koff);
    v2f b = *(const v2f*)(brow + j * 4 + koff);
    acc = __builtin_amdgcn_wmma_f32_16x16x4_f32(false, a, false, b,
                                                (short)0, acc, false, false);
  }
  const float bias = bout[ncol];
  const int mbase = (mtile << 4) + ((lane >> 4) << 3);
  #pragma unroll
  for (int i = 0; i < 8; ++i)
    Y[(size_t)(mbase + i) * DIN + ncol] = acc[i] + bias;
}

// ============================================================
// K7: bilinear 4x4 -> 32x32 blend of projected vectors, write decoded.
// align_corners=False half-pixel mapping; edge weights renormalize to
// clamp-to-edge (matches jax.image.resize for 2-tap upsample). 134MB store.
// ============================================================
__global__ __launch_bounds__(256) void k_upsample(const float* __restrict__ Y,
                                                  float* __restrict__ decoded) {
  const int w = blockIdx.x, h = blockIdx.y, b = blockIdx.z;
  const int t = threadIdx.x;            // 4 channels each
  float sy = (h + 0.5f) * 0.125f - 0.5f;
  float sx = (w + 0.5f) * 0.125f - 0.5f;
  float fy = sy - floorf(sy);
  float fx = sx - floorf(sx);
  int y0 = (int)floorf(sy), x0 = (int)floorf(sx);
  int y1 = y0 + 1, x1 = x0 + 1;
  y0 = y0 < 0 ? 0 : (y0 > 3 ? 3 : y0);
  y1 = y1 < 0 ? 0 : (y1 > 3 ? 3 : y1);
  x0 = x0 < 0 ? 0 : (x0 > 3 ? 3 : x0);
  x1 = x1 < 0 ? 0 : (x1 > 3 ? 3 : x1);
  const float4* __restrict__ Y4 = (const float4*)Y;
  size_t base = (size_t)b * 16;
  float4 r00 = Y4[(base + y0 * 4 + x0) * (DIN / 4) + t];
  float4 r01 = Y4[(base + y0 * 4 + x1) * (DIN / 4) + t];
  float4 r10 = Y4[(base + y1 * 4 + x0) * (DIN / 4) + t];
  float4 r11 = Y4[(base + y1 * 4 + x1) * (DIN / 4) + t];
  float w00 = (1.f - fy) * (1.f - fx), w01 = (1.f - fy) * fx;
  float w10 = fy * (1.f - fx),         w11 = fy * fx;
  float4 o;
  o.x = w00 * r00.x + w01 * r01.x + w10 * r10.x + w11 * r11.x;
  o.y = w00 * r00.y + w01 * r01.y + w10 * r10.y + w11 * r11.y;
  o.z = w00 * r00.z + w01 * r01.z + w10 * r10.z + w11 * r11.z;
  o.w = w00 * r00.w + w01 * r01.w + w10 * r10.w + w11 * r11.w;
  ((float4*)decoded)[((size_t)(b * HH + h) * WW + w) * (DIN / 4) + t] = o;
}

// ============================================================
extern "C" void kernel_launch(void* const* d_in, const int* in_sizes, int n_in,
                              void* d_out, int out_size, void* d_ws, size_t ws_size,
                              hipStream_t stream) {
  const float* t0    = (const float*)d_in[0];
  const float* t1    = (const float*)d_in[1];
  const float* noise = (const float*)d_in[2];
  const float* cb    = (const float*)d_in[3];
  const float* Win   = (const float*)d_in[4];
  const float* bin   = (const float*)d_in[5];
  const float* Wout  = (const float*)d_in[6];
  const float* bout  = (const float*)d_in[7];

  float* out = (float*)d_out;
  float* ws  = (float*)d_ws;

  float*    pooled = ws + WS_POOLED;
  float*    flat   = ws + WS_FLAT;
  float*    quant  = ws + WS_QUANT;
  float*    Y      = ws + WS_Y;
  float*    cnorm  = ws + WS_CNORM;
  unsigned* counts = (unsigned*)(ws + WS_COUNTS);
  int*      minidx = (int*)(ws + WS_MINIDX);

  float* out_decoded = out + OUT_DECODED;
  float* out_perp    = out + OUT_PERP;
  int*   out_used    = (int*)(out + OUT_USED);
  int*   out_idx     = (int*)(out + OUT_IDX);

  k_pool<<<dim3(16, 32), 256, 0, stream>>>(t0, t1, pooled);
  k_encode<<<128, 32, 0, stream>>>(pooled, Win, bin, flat);
  k_cnorm<<<KCB / 256, 256, 0, stream>>>(cb, cnorm, counts);
  k_argmin<<<NROWS / 16, 256, 0, stream>>>(flat, cb, cnorm, counts, minidx, out_idx);
  k_quant<<<NROWS, 32, 0, stream>>>(flat, cb, noise, minidx, quant);
  k_perp<<<1, 256, 0, stream>>>(counts, out_perp, out_used);
  k_decode_gemm<<<2048, 32, 0, stream>>>(quant, Wout, bout, Y);
  k_upsample<<<dim3(WW, HH, BB), 256, 0, stream>>>(Y, out_decoded);
}